// MessagePass_12463995093091
// MI455X (gfx1250) — compile-verified
//
#include <hip/hip_runtime.h>

typedef __attribute__((ext_vector_type(2))) float v2f;
typedef __attribute__((ext_vector_type(8))) float v8f;

#define N_NODES     50000
#define D           64
#define TWO_D       128
#define AGGR_ELEMS  (N_NODES * D)        // 3,200,000 floats
#define EDGES_PER_WG 128                 // 8 waves * 16 edges
#define WP_PITCH    160                  // dwords per packed k-pair row (bank-padded)

// ---------------------------------------------------------------------------
// Zero-fill the aggregation region of d_out (harness poisons it with 0xAA).
// ---------------------------------------------------------------------------
__global__ void zero_f4_kernel(float4* __restrict__ p, int n4) {
    int i = blockIdx.x * blockDim.x + threadIdx.x;
    if (i < n4) p[i] = make_float4(0.f, 0.f, 0.f, 0.f);
}

// ---------------------------------------------------------------------------
// Fused: m = relu([x_i|x_j] @ W + b); store m; atomic-scatter m into aggr.
// One wave computes a 16-edge x 64-col tile via V_WMMA_F32_16X16X4_F32.
// ---------------------------------------------------------------------------
__global__ __launch_bounds__(256) void msg_gemm_scatter_kernel(
    const float* __restrict__ x_i,
    const float* __restrict__ x_j,
    const int*   __restrict__ recipients,
    const float* __restrict__ W,     // [128, 64] row-major
    const float* __restrict__ bvec,  // [64]
    float* __restrict__ aggr,        // [N_NODES, 64]
    float* __restrict__ m_out,       // [E, 64]
    int n_edges)
{
    // W packed so (k, k+1) live in adjacent dwords: one ds_load_b64 per B frag.
    // Pitch 160 dwords per k-pair => the +2K half-wave lands on banks shifted
    // by 32, disjoint from lanes 0-15 -> conflict-free b64 LDS reads.
    __shared__ float Wlds[64 * WP_PITCH];   // 40 KB of the 320 KB WGP LDS

    const int tid = threadIdx.x;
    #pragma unroll
    for (int i = 0; i < (TWO_D * D) / 256; ++i) {   // 32 elements per thread
        int g = tid + i * 256;
        int k = g >> 6;          // 0..127
        int n = g & 63;          // 0..63
        Wlds[(k >> 1) * WP_PITCH + n * 2 + (k & 1)] = W[g];
    }
    __syncthreads();

    const int lane   = tid & 31;
    const int wave   = tid >> 5;        // 0..7
    const int lane15 = lane & 15;
    const int half   = lane >> 4;       // 0 or 1
    const int kOff   = half * 2;        // ISA A/B frag layout: upper half = K+2

    const int edgeBase = blockIdx.x * EDGES_PER_WG + wave * 16;
    if (edgeBase >= n_edges) return;    // wave-uniform: EXEC stays all ones

    const int rowA = edgeBase + lane15;
    const float* __restrict__ ai = x_i + (size_t)rowA * D;
    const float* __restrict__ aj = x_j + (size_t)rowA * D;

    v8f acc[4];
    #pragma unroll
    for (int nt = 0; nt < 4; ++nt) acc[nt] = (v8f){0.f,0.f,0.f,0.f,0.f,0.f,0.f,0.f};

    // K = 128 in 32 steps of 4. concat(x_i, x_j): k < 64 -> x_i, else x_j.
    // kOff in {0,2} so the b64 load never straddles the 64 boundary.
    #pragma unroll
    for (int ks = 0; ks < 32; ++ks) {
        const int kg = ks * 4 + kOff;                       // even, 0..126
        const float* src = (kg < 64) ? (ai + kg) : (aj + (kg - 64));
        v2f a = __builtin_nontemporal_load((const v2f*)src);  // streamed once

        const float* wrow = &Wlds[(kg >> 1) * WP_PITCH];
        v2f b0 = *(const v2f*)(wrow + (0 * 16 + lane15) * 2);
        v2f b1 = *(const v2f*)(wrow + (1 * 16 + lane15) * 2);
        v2f b2 = *(const v2f*)(wrow + (2 * 16 + lane15) * 2);
        v2f b3 = *(const v2f*)(wrow + (3 * 16 + lane15) * 2);

        acc[0] = __builtin_amdgcn_wmma_f32_16x16x4_f32(false, a, false, b0,
                                                       (short)0, acc[0], false, false);
        acc[1] = __builtin_amdgcn_wmma_f32_16x16x4_f32(false, a, false, b1,
                                                       (short)0, acc[1], false, false);
        acc[2] = __builtin_amdgcn_wmma_f32_16x16x4_f32(false, a, false, b2,
                                                       (short)0, acc[2], false, false);
        acc[3] = __builtin_amdgcn_wmma_f32_16x16x4_f32(false, a, false, b3,
                                                       (short)0, acc[3], false, false);
    }

    // Epilogue: bias + ReLU, NT-store m, atomic scatter-add into aggr.
    float bias[4];
    #pragma unroll
    for (int nt = 0; nt < 4; ++nt) bias[nt] = bvec[nt * 16 + lane15];

    // C/D layout: VGPR r, lanes 0-15 -> M = r, lanes 16-31 -> M = r + 8.
    const int rbase = edgeBase + 8 * half;
    #pragma unroll
    for (int r = 0; r < 8; ++r) {
        const int row = rbase + r;
        const int rec = recipients[row];
        float* mrow = m_out + (size_t)row * D + lane15;
        float* arow = aggr  + (size_t)rec * D + lane15;
        #pragma unroll
        for (int nt = 0; nt < 4; ++nt) {
            float v = acc[nt][r] + bias[nt];
            v = v > 0.f ? v : 0.f;
            __builtin_nontemporal_store(v, mrow + nt * 16);  // write-once stream
            atomicAdd(arow + nt * 16, v);                    // L2-resident, no-return
        }
    }
}

extern "C" void kernel_launch(void* const* d_in, const int* in_sizes, int n_in,
                              void* d_out, int out_size, void* d_ws, size_t ws_size,
                              hipStream_t stream) {
    const float* x_i  = (const float*)d_in[0];
    const float* x_j  = (const float*)d_in[1];
    const int*   rec  = (const int*)  d_in[2];
    const float* W    = (const float*)d_in[3];
    const float* bvec = (const float*)d_in[4];

    float* out  = (float*)d_out;
    float* aggr = out;                       // [50000, 64]
    float* m    = out + (size_t)AGGR_ELEMS;  // [E, 64]

    const int n_edges = in_sizes[0] / D;     // 800000

    // 1) zero the aggregation buffer
    const int n4 = AGGR_ELEMS / 4;
    zero_f4_kernel<<<(n4 + 255) / 256, 256, 0, stream>>>((float4*)aggr, n4);

    // 2) fused GEMM + ReLU + scatter (800000 % 128 == 0 -> no tail)
    const int blocks = (n_edges + EDGES_PER_WG - 1) / EDGES_PER_WG;
    msg_gemm_scatter_kernel<<<blocks, 256, 0, stream>>>(
        x_i, x_j, rec, W, bvec, aggr, m, n_edges);
}